// MemN2N_SS_3813930959278
// MI455X (gfx1250) — compile-verified
//
#include <hip/hip_runtime.h>

#define BB 128
#define SS 200
#define LL 50
#define QQ 20
#define VV 50000
#define DD 128
#define KK 3

typedef __attribute__((ext_vector_type(2))) float v2f;
typedef __attribute__((ext_vector_type(8))) float v8f;

// position encoding l_kj = (1 - j/J) - (k/d)(1 - 2j/J), 1-indexed
__device__ __forceinline__ float pe_coef(int j0, int J, float kf) {
    float jr = (float)(j0 + 1) / (float)J;
    return (1.0f - jr) - kf * (1.0f - 2.0f * jr);
}

// u[b,d] = sum_q embs0[query[b,q], d] * pe_q(q,d)
__global__ void query_embed_kernel(const int* __restrict__ query,
                                   const float* __restrict__ emb0,
                                   float* __restrict__ u) {
    int b = blockIdx.x, d = threadIdx.x;
    __shared__ int qidx[QQ];
    if (d < QQ) qidx[d] = query[b * QQ + d];
    __syncthreads();
    float kf = (float)(d + 1) / (float)DD;
    float acc = 0.f;
#pragma unroll
    for (int q = 0; q < QQ; ++q)
        acc += emb0[(size_t)qidx[q] * DD + d] * pe_coef(q, QQ, kf);
    u[b * DD + d] = acc;
}

// bag[e,b,s,d] = sum_l embs[e][story[b,s,l], d] * pe_s(l,d)
// One block per (e,b,s); tables are L2-resident (4 x 25.6 MB < 192 MB L2).
__global__ void story_bag_kernel(const int* __restrict__ story,
                                 const float* __restrict__ embs,
                                 float* __restrict__ bag) {
    int lin = blockIdx.x;
    int e   = lin / (BB * SS);
    int rem = lin - e * (BB * SS);
    int b   = rem / SS;
    int s   = rem - b * SS;
    int d   = threadIdx.x;
    const float* tab = embs + (size_t)e * VV * DD;
    const int* row = story + ((size_t)b * SS + s) * LL;
    __shared__ int idx[LL];
    if (d < LL) idx[d] = row[d];
    __syncthreads();
    float kf = (float)(d + 1) / (float)DD;
    float acc = 0.f;
#pragma unroll 5
    for (int l = 0; l < LL; ++l)
        acc += tab[(size_t)idx[l] * DD + d] * pe_coef(l, LL, kf);
    bag[(((size_t)e * BB + b) * SS + s) * DD + d] = acc;
}

// One block per batch b: scores = (bag[hop]+TA[hop]) . u, softmax over S,
// o = p . (bag[hop+1]+TC[hop]), u += o.  128 threads = 4 wave32.
__global__ void hop_kernel(int hop,
                           const float* __restrict__ bag,
                           const float* __restrict__ TA,
                           const float* __restrict__ TC,
                           float* __restrict__ u) {
    __shared__ float u_s[DD];
    __shared__ float sc[SS];
    __shared__ float red[DD];
    int b = blockIdx.x, tid = threadIdx.x;
    u_s[tid] = u[b * DD + tid];
    __syncthreads();

    int wv = tid >> 5, lane = tid & 31;
    // scores: wave w handles slots s = w, w+4, ...
    for (int s = wv; s < SS; s += 4) {
        const float* mrow  = bag + (((size_t)hop * BB + b) * SS + s) * DD;
        const float* tarow = TA + ((size_t)hop * SS + s) * DD;
        float part = 0.f;
#pragma unroll
        for (int dd = lane; dd < DD; dd += 32)
            part += (mrow[dd] + tarow[dd]) * u_s[dd];
#pragma unroll
        for (int off = 16; off >= 1; off >>= 1)
            part += __shfl_xor(part, off, 32);
        if (lane == 0) sc[s] = part;
    }
    __syncthreads();

    // softmax over S=200 in LDS
    float mx = -3.0e38f;
    for (int s = tid; s < SS; s += DD) mx = fmaxf(mx, sc[s]);
    red[tid] = mx;
    __syncthreads();
    for (int st = 64; st >= 1; st >>= 1) {
        if (tid < st) red[tid] = fmaxf(red[tid], red[tid + st]);
        __syncthreads();
    }
    float gmax = red[0];
    __syncthreads();
    float ls = 0.f;
    for (int s = tid; s < SS; s += DD) {
        float e = __expf(sc[s] - gmax);
        sc[s] = e;
        ls += e;
    }
    red[tid] = ls;
    __syncthreads();
    for (int st = 64; st >= 1; st >>= 1) {
        if (tid < st) red[tid] += red[tid + st];
        __syncthreads();
    }
    float inv = 1.0f / red[0];
    __syncthreads();

    // o[d] = sum_s p[s] * (bag[hop+1][b,s,d] + TC[hop,s,d]);  u += o
    const float* crow0  = bag + (((size_t)(hop + 1) * BB + b) * SS) * DD;
    const float* tcrow0 = TC + ((size_t)hop * SS) * DD;
    float acc = 0.f;
    for (int s = 0; s < SS; ++s)
        acc += sc[s] * (crow0[(size_t)s * DD + tid] + tcrow0[(size_t)s * DD + tid]);
    u[b * DD + tid] = u_s[tid] + acc * inv;
}

// out[b,v] = sum_d u[b,d] * W[v,d]   (W = embs[K], row-major (V,D))
// Full-precision f32 WMMA 16x16x4. One block per 16-col tile; wave w -> 16-row tile w.
// A 16x4 layout: lanes 0-15 (M=lane): VGPR0=K0,VGPR1=K1; lanes 16-31: K2,K3.
// C/D layout: VGPR r: lanes 0-15 -> (M=r, N=lane); lanes 16-31 -> (M=r+8, N=lane-16).
__global__ void final_gemm_wmma(const float* __restrict__ u,
                                const float* __restrict__ W,
                                float* __restrict__ out) {
    int lane = threadIdx.x & 31;
    int wv   = threadIdx.x >> 5;
    int row0 = wv * 16;
    int col0 = blockIdx.x * 16;
    int mn = lane & 15;   // M for A-frag, N for B-frag
    int hi = lane >> 4;   // selects K offset 0 or 2 within each K=4 step

    const float* arow = u + (size_t)(row0 + mn) * DD + 2 * hi;
    const float* brow = W + (size_t)(col0 + mn) * DD + 2 * hi;

    v8f acc = {0.f, 0.f, 0.f, 0.f, 0.f, 0.f, 0.f, 0.f};
#pragma unroll 8
    for (int k = 0; k < DD; k += 4) {
        v2f a  = *(const v2f*)(arow + k);
        v2f bf = *(const v2f*)(brow + k);
        acc = __builtin_amdgcn_wmma_f32_16x16x4_f32(
            /*neg_a=*/false, a, /*neg_b=*/false, bf,
            /*c_mod=*/(short)0, acc, /*reuse_a=*/false, /*reuse_b=*/false);
    }
#pragma unroll
    for (int r = 0; r < 8; ++r) {
        int m = r + 8 * hi;
        out[(size_t)(row0 + m) * VV + (col0 + mn)] = acc[r];
    }
}

extern "C" void kernel_launch(void* const* d_in, const int* in_sizes, int n_in,
                              void* d_out, int out_size, void* d_ws, size_t ws_size,
                              hipStream_t stream) {
    const int*   story = (const int*)d_in[0];    // (B,S,L)
    const int*   query = (const int*)d_in[1];    // (B,Q)
    const float* embs  = (const float*)d_in[2];  // (K+1,V,D)
    const float* TA    = (const float*)d_in[3];  // (K,S,D)
    const float* TC    = (const float*)d_in[4];  // (K,S,D)
    float* out = (float*)d_out;                  // (B,V)

    float* bag = (float*)d_ws;                       // 4*B*S*D floats
    float* u   = bag + (size_t)4 * BB * SS * DD;     // B*D floats

    query_embed_kernel<<<BB, DD, 0, stream>>>(query, embs, u);
    story_bag_kernel<<<4 * BB * SS, DD, 0, stream>>>(story, embs, bag);
    for (int hop = 0; hop < KK; ++hop)
        hop_kernel<<<BB, DD, 0, stream>>>(hop, bag, TA, TC, u);
    final_gemm_wmma<<<VV / 16, 256, 0, stream>>>(
        u, embs + (size_t)KK * VV * DD, out);
}